// CascadeRefineHead_74938589381109
// MI455X (gfx1250) — compile-verified
//
#include <hip/hip_runtime.h>

// CascadeRefineHead for MI455X (gfx1250, wave32).
// fp32 end-to-end using V_WMMA_F32_16X16X4_F32 for all three GEMM stages.
// CDNA5 paths: v_wmma f32, global_load_async_to_lds_b32 + s_wait_asynccnt,
// ds_bpermute lane exchange, LDS-coalesced float4 output stores.

typedef __attribute__((ext_vector_type(2))) float v2f;
typedef __attribute__((ext_vector_type(8))) float v8f;
typedef __attribute__((address_space(3))) float lds_float;

#define WMMA_F32_16x16x4(a, b, c) \
    __builtin_amdgcn_wmma_f32_16x16x4_f32(false, (a), false, (b), (short)0, (c), false, false)

__device__ __forceinline__ unsigned crh_lds_addr(float* p) {
    // generic (LDS-backed) pointer -> 32-bit LDS byte offset for async-LDS VDST
    return (unsigned)(unsigned long long)(lds_float*)p;
}

// ---------------------------------------------------------------------------
// Pad kernel: zero-padded 80-row weights so WMMA A loads are unconditional.
// ---------------------------------------------------------------------------
__global__ __launch_bounds__(256)
void crh_pad_kernel(const float* __restrict__ aw, const float* __restrict__ ab,
                    const float* __restrict__ ow, const float* __restrict__ ob,
                    const float* __restrict__ w,  const float* __restrict__ b,
                    float* __restrict__ adjW, float* __restrict__ adjB,
                    float* __restrict__ offW, float* __restrict__ offB,
                    float* __restrict__ wP,   float* __restrict__ bP)
{
    const int i = blockIdx.x * 256 + threadIdx.x;
    if (i < 80 * 64) {
        int r = i / 64, c = i - r * 64;
        adjW[i] = (r < 72) ? aw[r * 64 + c] : 0.0f;
    }
    if (i < 80 * 72) {
        int r = i / 72, c = i - r * 72;
        offW[i] = (r < 72) ? ow[r * 72 + c] : 0.0f;
        wP[i]   = (r < 72) ? w[r * 72 + c]  : 0.0f;
    }
    if (i < 80) {
        adjB[i] = (i < 72) ? ab[i] : 0.0f;
        offB[i] = (i < 72) ? ob[i] : 0.0f;
        bP[i]   = (i < 72) ? b[i]  : 0.0f;
    }
}

// ---------------------------------------------------------------------------
// Kernel A (templated on HW so every stride is a compile-time immediate):
// fa = adjust_w(72x64) @ f + b per 16-pixel tile (one wave). Level 0 also
// computes offs = offset_w(72x72) @ fa + b via an LDS D->B relayout.
// ---------------------------------------------------------------------------
template<int HW, bool DO_OFF>
__global__ __launch_bounds__(32)
void crh_adjust_kernel(const float* __restrict__ f,
                       const float* __restrict__ adjW,   // 80x64, zero-padded
                       const float* __restrict__ adjB,   // 80
                       const float* __restrict__ offW,   // 80x72, zero-padded
                       const float* __restrict__ offB,   // 80
                       float* __restrict__ fa,
                       float* __restrict__ offs)
{
    __shared__ float lds[80 * 17];   // stride 17 -> conflict-free scatter/reload

    const int lane = threadIdx.x;
    const int half = lane >> 4;
    const int lm   = lane & 15;

    const int pix0 = blockIdx.x * 16;        // uniform
    const int b    = pix0 / HW;              // uniform (16 divides HW=16000/4000, not 1000)
    const int hw   = (pix0 - b * HW) + lm;

    // per-lane bases with the lane-dependent parts folded in once
    const float* fbh = f    + (b * 64 + 2 * half) * HW + hw;
    const float* awl = adjW + lm * 64 + 2 * half;
    const float* abl = adjB + 8 * half;

    v8f acc[5];
#pragma unroll
    for (int m = 0; m < 5; ++m)
#pragma unroll
        for (int v = 0; v < 8; ++v)
            acc[m][v] = abl[16 * m + v];

    // GEMM1: K = 64 -> 16 WMMA steps of K=4; all loads immediate-offset.
#pragma unroll 4
    for (int k = 0; k < 16; ++k) {
        v2f bv;
        bv.x = fbh[(4 * k) * HW];
        bv.y = fbh[(4 * k + 1) * HW];
#pragma unroll
        for (int m = 0; m < 5; ++m) {
            v2f av;
            av.x = awl[m * 1024 + 4 * k];
            av.y = awl[m * 1024 + 4 * k + 1];
            acc[m] = WMMA_F32_16x16x4(av, bv, acc[m]);
        }
    }

    // fa store: m=0..3 unconditional, m=4 only channels 64..71 (half==0).
    float* fo = fa + (b * 72 + 8 * half) * HW + hw;
#pragma unroll
    for (int m = 0; m < 4; ++m)
#pragma unroll
        for (int v = 0; v < 8; ++v)
            fo[(16 * m + v) * HW] = acc[m][v];
    if (half == 0) {
#pragma unroll
        for (int v = 0; v < 8; ++v)
            fo[(64 + v) * HW] = acc[4][v];
    }

    if (DO_OFF) {
        // stage D-layout tile into LDS (rows = channels, stride 17)
        float* lh = lds + (8 * half) * 17 + lm;
#pragma unroll
        for (int m = 0; m < 5; ++m)
#pragma unroll
            for (int v = 0; v < 8; ++v)
                lh[(16 * m + v) * 17] = acc[m][v];
        __syncthreads();

        v8f acc2[5];
        const float* obl = offB + 8 * half;
#pragma unroll
        for (int m = 0; m < 5; ++m)
#pragma unroll
            for (int v = 0; v < 8; ++v)
                acc2[m][v] = obl[16 * m + v];

        const float* owl = offW + lm * 72 + 2 * half;
        const float* lb  = lds + (2 * half) * 17 + lm;
#pragma unroll 3
        for (int k = 0; k < 18; ++k) {
            v2f bv;
            bv.x = lb[(4 * k) * 17];
            bv.y = lb[(4 * k + 1) * 17];
#pragma unroll
            for (int m = 0; m < 5; ++m) {
                v2f av;
                av.x = owl[m * 1152 + 4 * k];
                av.y = owl[m * 1152 + 4 * k + 1];
                acc2[m] = WMMA_F32_16x16x4(av, bv, acc2[m]);
            }
        }

        float* oo = offs + (b * 72 + 8 * half) * HW + hw;
#pragma unroll
        for (int m = 0; m < 4; ++m)
#pragma unroll
            for (int v = 0; v < 8; ++v)
                oo[(16 * m + v) * HW] = acc2[m][v];
        if (half == 0) {
#pragma unroll
            for (int v = 0; v < 8; ++v)
                oo[(64 + v) * HW] = acc2[4][v];
        }
    }
}

// ---------------------------------------------------------------------------
// Kernel B helper: one pyramid level for one wave's 16 points.
// Gather 72ch x 16pt of fa[l] into the wave's LDS panel (B-layout, stride 17),
// WMMA-project (A = LDS weight, stride 73), then relayout D through the same
// panel as [point][channel] rows (stride 76) and emit coalesced float4 stores.
// ---------------------------------------------------------------------------
template<int Hl, int Wl>
__device__ __forceinline__ void crh_level(const float* __restrict__ fab,  // fa + b*72*HWl
                                          float* __restrict__ sU,        // per-wave panel (1224 f)
                                          const float* __restrict__ sW,  // 80x73 LDS weight
                                          const float* __restrict__ sBias,
                                          float* __restrict__ outw,      // wave's output slab
                                          int lane, int half, int lm,
                                          float gx, float gy, float dx, float dy)
{
    constexpr int HWl = Hl * Wl;

    float x = (gx + 1.0f) * 0.5f * (float)(Wl - 1) + dx;
    float y = (gy + 1.0f) * 0.5f * (float)(Hl - 1) + dy;
    x = fminf(fmaxf(x, 0.0f), (float)(Wl - 1));
    y = fminf(fmaxf(y, 0.0f), (float)(Hl - 1));
    float x0f = floorf(x), y0f = floorf(y);
    int x0 = (int)x0f, y0 = (int)y0f;
    int x1 = min(x0 + 1, Wl - 1), y1 = min(y0 + 1, Hl - 1);
    float wx = x - x0f, wy = y - y0f;
    float w00 = (1.0f - wx) * (1.0f - wy), w01 = wx * (1.0f - wy);
    float w10 = (1.0f - wx) * wy,          w11 = wx * wy;

    // 4 per-lane tap base pointers; all 36 channel-pair loads use immediates.
    const float* bh  = fab + half * HWl;
    const float* c00 = bh + y0 * Wl + x0;
    const float* c01 = bh + y0 * Wl + x1;
    const float* c10 = bh + y1 * Wl + x0;
    const float* c11 = bh + y1 * Wl + x1;

    float* su = sU + half * 17 + lm;   // row (2i+half), col lm
#pragma unroll 6
    for (int i = 0; i < 36; ++i) {
        float v = w00 * c00[2 * i * HWl] + w01 * c01[2 * i * HWl]
                + w10 * c10[2 * i * HWl] + w11 * c11[2 * i * HWl];
        su[(2 * i) * 17] = v;
    }

    // out(72x16) = weight @ panel + bias, all operands in LDS.
    v8f acc[5];
    const float* sbb = sBias + 8 * half;
#pragma unroll
    for (int m = 0; m < 5; ++m)
#pragma unroll
        for (int v = 0; v < 8; ++v)
            acc[m][v] = sbb[16 * m + v];

    const float* swl = sW + lm * 73 + 2 * half;
    const float* sbk = sU + (2 * half) * 17 + lm;
#pragma unroll 3
    for (int k = 0; k < 18; ++k) {
        v2f bv;
        bv.x = sbk[(4 * k) * 17];
        bv.y = sbk[(4 * k + 1) * 17];
#pragma unroll
        for (int m = 0; m < 5; ++m) {
            v2f av;
            av.x = swl[m * 1168 + 4 * k];        // 16*73 = 1168
            av.y = swl[m * 1168 + 4 * k + 1];
            acc[m] = WMMA_F32_16x16x4(av, bv, acc[m]);
        }
    }

    // Relayout D -> [point][channel] rows (stride 76, 16B-aligned) in the
    // same panel (all B reads are already done; per-wave LDS is in-order).
    float* so = sU + lm * 76 + 8 * half;
#pragma unroll
    for (int m = 0; m < 4; ++m)
#pragma unroll
        for (int v = 0; v < 8; ++v)
            so[16 * m + v] = acc[m][v];
    if (half == 0) {
#pragma unroll
        for (int v = 0; v < 8; ++v)
            so[64 + v] = acc[4][v];
    }

    // Coalesced output: 16 points x 72 floats = 288 float4 = 9 per lane.
    float4* o4 = (float4*)outw;
#pragma unroll
    for (int t = 0; t < 9; ++t) {
        int flat = t * 32 + lane;         // = pt*18 + c4
        int pt = flat / 18;
        int c4 = flat - pt * 18;
        float4 val = *(const float4*)(sU + pt * 76 + c4 * 4);
        o4[flat] = val;
    }
}

__global__ __launch_bounds__(256)
void crh_sample_kernel(const float* __restrict__ grid,
                       const float* __restrict__ offs,
                       const float* __restrict__ fa0,
                       const float* __restrict__ fa1,
                       const float* __restrict__ fa2,
                       const float* __restrict__ wPad,   // 80x72, zero-padded
                       const float* __restrict__ bPad,   // 80
                       float* __restrict__ out)
{
    __shared__ __align__(16) float sW[80 * 73];
    __shared__ float sBias[80];
    __shared__ __align__(16) float sPanel[8][1224];   // per-wave gather/output panel

    const int tid  = threadIdx.x;
    const int wv   = tid >> 5;
    const int lane = tid & 31;
    const int half = lane >> 4;
    const int lm   = lane & 15;

    // --- async-DMA padded weight + bias into LDS (ASYNCcnt path) ---
    for (int idx = tid; idx < 80 * 72; idx += 256) {
        int r = idx / 72, c = idx - r * 72;
        unsigned ldsb = crh_lds_addr(&sW[r * 73 + c]);
        asm volatile("global_load_async_to_lds_b32 %0, %1, %2"
                     :: "v"(ldsb), "v"((unsigned)idx * 4u), "s"(wPad) : "memory");
    }
    if (tid < 80) {
        unsigned ldsb = crh_lds_addr(&sBias[tid]);
        asm volatile("global_load_async_to_lds_b32 %0, %1, %2"
                     :: "v"(ldsb), "v"((unsigned)tid * 4u), "s"(bPad) : "memory");
    }
    asm volatile("s_wait_asynccnt 0" ::: "memory");
    __syncthreads();

    const int b    = blockIdx.x / 216;                       // uniform (128 | 27648)
    const int rem0 = (blockIdx.x - b * 216) * 128 + wv * 16; // wave-uniform
    const int pt   = b * 27648 + rem0 + lm;
    const int s    = (rem0 + lm) % 36;

    const float gx = grid[pt * 2 + 0];
    const float gy = grid[pt * 2 + 1];

    // Offset sample at level-0 res; lanes 0-15 -> dy (ch 2s), 16-31 -> dx.
    float val;
    {
        float ox = (gx + 1.0f) * 0.5f * 199.0f;
        float oy = (gy + 1.0f) * 0.5f * 79.0f;
        const float* img = offs + (b * 72 + 2 * s + half) * 16000;
        float xx = fminf(fmaxf(ox, 0.0f), 199.0f);
        float yy = fminf(fmaxf(oy, 0.0f), 79.0f);
        float x0f = floorf(xx), y0f = floorf(yy);
        int x0 = (int)x0f, y0 = (int)y0f;
        int x1 = min(x0 + 1, 199), y1 = min(y0 + 1, 79);
        float fx = xx - x0f, fy = yy - y0f;
        float v00 = img[y0 * 200 + x0], v01 = img[y0 * 200 + x1];
        float v10 = img[y1 * 200 + x0], v11 = img[y1 * 200 + x1];
        val = (v00 * (1.0f - fx) + v01 * fx) * (1.0f - fy)
            + (v10 * (1.0f - fx) + v11 * fx) * fy;
    }
    const float dy = __shfl(val, lm, 32);
    const float dx = __shfl(val, lm + 16, 32);

    float* su = sPanel[wv];
    const int obl = (b * 3) * 27648 + rem0;   // (b*3+l)*27648 + rem0, l folded below

    crh_level<80, 200>(fa0 + (b * 72) * 16000, su, sW, sBias,
                       out + (size_t)(obl + 0 * 27648) * 72,
                       lane, half, lm, gx, gy, dx, dy);
    crh_level<40, 100>(fa1 + (b * 72) * 4000, su, sW, sBias,
                       out + (size_t)(obl + 1 * 27648) * 72,
                       lane, half, lm, gx, gy, dx, dy);
    crh_level<20, 50>(fa2 + (b * 72) * 1000, su, sW, sBias,
                      out + (size_t)(obl + 2 * 27648) * 72,
                      lane, half, lm, gx, gy, dx, dy);
}

// ---------------------------------------------------------------------------
// Launcher
// ---------------------------------------------------------------------------
extern "C" void kernel_launch(void* const* d_in, const int* in_sizes, int n_in,
                              void* d_out, int out_size, void* d_ws, size_t ws_size,
                              hipStream_t stream)
{
    const float* f0       = (const float*)d_in[0];
    const float* f1       = (const float*)d_in[1];
    const float* f2       = (const float*)d_in[2];
    const float* grid     = (const float*)d_in[3];
    const float* adjust_w = (const float*)d_in[4];
    const float* adjust_b = (const float*)d_in[5];
    const float* offset_w = (const float*)d_in[6];
    const float* offset_b = (const float*)d_in[7];
    const float* weight   = (const float*)d_in[8];
    const float* bias     = (const float*)d_in[9];
    float* out = (float*)d_out;

    // Workspace: padded params | fa0 | fa1 | fa2 | offsets
    float* adjW = (float*)d_ws;                  // 80*64
    float* adjB = adjW + 80 * 64;                // 80
    float* offW = adjB + 80;                     // 80*72
    float* offB = offW + 80 * 72;                // 80
    float* wP   = offB + 80;                     // 80*72
    float* bP   = wP + 80 * 72;                  // 80
    float* fa0  = bP + 80;                       // 16*72*16000
    float* fa1  = fa0 + (size_t)16 * 72 * 16000; // 16*72*4000
    float* fa2  = fa1 + (size_t)16 * 72 * 4000;  // 16*72*1000
    float* offz = fa2 + (size_t)16 * 72 * 1000;  // 16*72*16000

    crh_pad_kernel<<<(80 * 72 + 255) / 256, 256, 0, stream>>>(
        adjust_w, adjust_b, offset_w, offset_b, weight, bias,
        adjW, adjB, offW, offB, wP, bP);

    crh_adjust_kernel<16000, true><<<16 * 16000 / 16, 32, 0, stream>>>(
        f0, adjW, adjB, offW, offB, fa0, offz);
    crh_adjust_kernel<4000, false><<<16 * 4000 / 16, 32, 0, stream>>>(
        f1, adjW, adjB, offW, offB, fa1, offz);
    crh_adjust_kernel<1000, false><<<16 * 1000 / 16, 32, 0, stream>>>(
        f2, adjW, adjB, offW, offB, fa2, offz);

    const int npts = 16 * 768 * 36;              // 442368 = 3456 * 128
    crh_sample_kernel<<<npts / 128, 256, 0, stream>>>(
        grid, offz, fa0, fa1, fa2, wP, bP, out);
}